// HGTLayer_43207370998333
// MI455X (gfx1250) — compile-verified
//
#include <hip/hip_runtime.h>
#include <math.h>
#include <stdint.h>

typedef float v2f __attribute__((ext_vector_type(2)));
typedef float v8f __attribute__((ext_vector_type(8)));

#define DIM 256
#define NHEAD 8
#define DHEAD 32

// ---------------------------------------------------------------------------
// helpers
// ---------------------------------------------------------------------------
__device__ __forceinline__ unsigned encodeOrdered(float f) {
    unsigned u = __float_as_uint(f);
    return (u & 0x80000000u) ? ~u : (u | 0x80000000u);
}
__device__ __forceinline__ float decodeOrdered(unsigned u) {
    unsigned b = (u & 0x80000000u) ? (u ^ 0x80000000u) : ~u;
    return __uint_as_float(b);
}

// ---------------------------------------------------------------------------
// zero fill
// ---------------------------------------------------------------------------
__global__ void zero_kernel(float* __restrict__ p, long long n) {
    long long i = (long long)blockIdx.x * blockDim.x + threadIdx.x;
    if (i < n) p[i] = 0.0f;
}

// ---------------------------------------------------------------------------
// Fold per-head 32x32 matrix (Watt or Wmsg) into a 256x256 projection weight:
// outF[h*32+f][j] = sum_d Whead[d][f] * Wsrc[h*32+d][j]
// ---------------------------------------------------------------------------
__global__ __launch_bounds__(256)
void fold_head_kernel(const float* __restrict__ Wsrc, const float* __restrict__ Whead,
                      float* __restrict__ outF) {
    int o = blockIdx.x;          // 0..255  (h*32+f)
    int j = threadIdx.x;         // 0..255
    int h = o >> 5, f = o & 31;
    float acc = 0.0f;
#pragma unroll
    for (int d = 0; d < 32; ++d)
        acc = fmaf(Whead[d * 32 + f], Wsrc[(h * 32 + d) * DIM + j], acc);
    outF[o * DIM + j] = acc;
}

// ---------------------------------------------------------------------------
// Fold the RTE linear into the folded projection:
// outM[o][i] = sum_j WF[o][j] * rW[j][i]    (rW is torch [out,in] = [j][i])
// outb[o]    = sum_j WF[o][j] * rb[j]
// ---------------------------------------------------------------------------
__global__ __launch_bounds__(256)
void fold_rte_kernel(const float* __restrict__ WF, const float* __restrict__ rW,
                     const float* __restrict__ rb,
                     float* __restrict__ outM, float* __restrict__ outb) {
    int o = blockIdx.x;
    int i = threadIdx.x;
    float acc = 0.0f;
    for (int j = 0; j < DIM; ++j)
        acc = fmaf(WF[o * DIM + j], rW[j * DIM + i], acc);
    outM[o * DIM + i] = acc;
    if (i == 0) {
        float ab = 0.0f;
        for (int j = 0; j < DIM; ++j) ab = fmaf(WF[o * DIM + j], rb[j], ab);
        outb[o] = ab;
    }
}

// ---------------------------------------------------------------------------
// fp32 WMMA GEMM:  OUT[r, n] = sum_j IN[r, j] * W[n, j] (+ bias[n])
// IN: [rows, 256]  W: [256, 256] row-major (torch Linear [out,in])
// One 512-thread block (16 waves) computes a 32-row x 256-col panel.
// Each wave owns one N tile and TWO 16x16 M tiles -> 2 v_wmma per B fetch.
// A panel staged global->LDS via GLOBAL_LOAD_ASYNC_TO_LDS_B128 (ASYNCcnt)
// for full panels; ragged tail blocks use a guarded zero-fill path.
// ---------------------------------------------------------------------------
__global__ __launch_bounds__(512)
void gemm256_wmma_kernel(const float* __restrict__ IN, const float* __restrict__ W,
                         const float* __restrict__ bias, float* __restrict__ OUT,
                         int rows) {
    __shared__ float Atile[32][260];   // pad 256->260 words: row start stays 16B
                                       // aligned (260*4 = 65*16) and column
                                       // reads hit distinct banks
    const int m0  = blockIdx.x * 32;
    const int tid = threadIdx.x;

    if (m0 + 32 <= rows) {
        // CDNA5 async path: 32x256 floats = 2048 16-byte chunks, 4 per thread.
        const unsigned ldsbase = (unsigned)(uintptr_t)(&Atile[0][0]);
#pragma unroll
        for (int q = 0; q < 4; ++q) {
            const int chunk = tid + q * 512;
            const int r  = chunk >> 6;          // 64 chunks per row
            const int c4 = (chunk & 63) << 2;   // starting float within row
            const unsigned lds = ldsbase + (unsigned)(r * 260 + c4) * 4u;
            const unsigned long long ga =
                (unsigned long long)(uintptr_t)(IN + (size_t)(m0 + r) * DIM + c4);
            asm volatile("global_load_async_to_lds_b128 %0, %1, off"
                         :: "v"(lds), "v"(ga) : "memory");
        }
        asm volatile("s_wait_asynccnt 0" ::: "memory");
    } else {
        // tail panel: guarded, zero-filled
        for (int i = tid; i < 32 * DIM; i += 512) {
            int r = i >> 8, c = i & 255;
            int gr = m0 + r;
            Atile[r][c] = (gr < rows) ? IN[(size_t)gr * DIM + c] : 0.0f;
        }
    }
    __syncthreads();

    const int wave  = tid >> 5;        // 0..15 -> N tile
    const int lane  = tid & 31;
    const int n0    = wave * 16;
    const int nrow  = lane & 15;       // A row within tile; also B column
    const int khalf = (lane >> 4) * 2; // lanes 0-15: K 0,1 ; lanes 16-31: K 2,3

    v8f acc0 = {};
    v8f acc1 = {};
#pragma unroll 4
    for (int kk = 0; kk < DIM; kk += 4) {
        v2f b  = *(const v2f*)&W[(size_t)(n0 + nrow) * DIM + kk + khalf];
        v2f a0 = *(const v2f*)&Atile[nrow][kk + khalf];
        v2f a1 = *(const v2f*)&Atile[16 + nrow][kk + khalf];
        acc0 = __builtin_amdgcn_wmma_f32_16x16x4_f32(
            false, a0, false, b, (short)0, acc0, false, false);
        acc1 = __builtin_amdgcn_wmma_f32_16x16x4_f32(
            false, a1, false, b, (short)0, acc1, false, false);
    }

    // D layout: VGPR i -> M = i + (lane>=16 ? 8 : 0), N = lane&15
    const int nb    = lane & 15;
    const int mbase = (lane >> 4) * 8;
    const float badd = bias ? bias[n0 + nb] : 0.0f;
#pragma unroll
    for (int i = 0; i < 8; ++i) {
        int m = m0 + mbase + i;
        if (m < rows) OUT[(size_t)m * DIM + n0 + nb] = acc0[i] + badd;
        int m2 = m + 16;
        if (m2 < rows) OUT[(size_t)m2 * DIM + n0 + nb] = acc1[i] + badd;
    }
}

// ---------------------------------------------------------------------------
// Edge pass 1: attention logits + running segment max.
// One wave per edge; lane = feature within head; loop over 8 heads.
// att[e,h] = < KW[src,h,:] + KR[tau,h,:], Q[dst,h,:] > * prior/sqrt(32)
// ---------------------------------------------------------------------------
__global__ __launch_bounds__(256)
void edge_att_kernel(const float* __restrict__ KW, const float* __restrict__ KR,
                     const float* __restrict__ Q,
                     const int* __restrict__ src, const int* __restrict__ dst,
                     const int* __restrict__ tt,
                     float* __restrict__ att, unsigned* __restrict__ amax,
                     const float* __restrict__ prior, int etype, int maxlen, int ne) {
    const int lane = threadIdx.x & 31;
    const int e = blockIdx.x * 8 + (threadIdx.x >> 5);
    if (e >= ne) return;
    const int s   = src[e];
    const int d   = dst[e];
    const int tau = maxlen - tt[e];
    const float scale = prior[etype] * 0.17677669529663687f;   // 1/sqrt(32)
    const float* kwp = KW + (size_t)s * DIM;
    const float* krp = KR + (size_t)tau * DIM;
    const float* qp  = Q  + (size_t)d * DIM;
#pragma unroll
    for (int h = 0; h < NHEAD; ++h) {
        int idx = h * DHEAD + lane;
        float v = (kwp[idx] + krp[idx]) * qp[idx];
        for (int off = 16; off > 0; off >>= 1) v += __shfl_xor(v, off, 32);
        if (lane == 0) {
            float a = v * scale;
            att[(size_t)e * NHEAD + h] = a;
            atomicMax(&amax[(size_t)d * NHEAD + h], encodeOrdered(a));
        }
    }
}

// ---------------------------------------------------------------------------
// Edge pass 2: ex = exp(att - amax[dst]); att <- ex; den[dst] += ex
// One thread per (edge, head).
// ---------------------------------------------------------------------------
__global__ __launch_bounds__(256)
void edge_softmax_kernel(const int* __restrict__ dst, const unsigned* __restrict__ amax,
                         float* __restrict__ att, float* __restrict__ den, int ne) {
    long long i = (long long)blockIdx.x * blockDim.x + threadIdx.x;
    if (i >= (long long)ne * NHEAD) return;
    int e = (int)(i >> 3), h = (int)(i & 7);
    int d = dst[e];
    float m  = decodeOrdered(amax[(size_t)d * NHEAD + h]);
    float ex = __expf(att[i] - m);
    att[i] = ex;
    atomicAdd(&den[(size_t)d * NHEAD + h], ex);
}

// ---------------------------------------------------------------------------
// Edge pass 3: agg[dst] += (ex/den[dst]) * (MW[src] + MR[tau])
// One 256-thread block per edge (thread = h*32+f).
// ---------------------------------------------------------------------------
__global__ __launch_bounds__(256)
void edge_scatter_kernel(const float* __restrict__ MW, const float* __restrict__ MR,
                         const int* __restrict__ src, const int* __restrict__ dst,
                         const int* __restrict__ tt,
                         const float* __restrict__ att, const float* __restrict__ den,
                         float* __restrict__ agg, int maxlen, int ne) {
    const int e = blockIdx.x;
    if (e >= ne) return;
    const int tid = threadIdx.x;
    const int h   = tid >> 5;
    const int s   = src[e];
    const int d   = dst[e];
    const int tau = maxlen - tt[e];
    const float w = att[(size_t)e * NHEAD + h] / den[(size_t)d * NHEAD + h];
    const float msg = MW[(size_t)s * DIM + tid] + MR[(size_t)tau * DIM + tid];
    atomicAdd(&agg[(size_t)d * DIM + tid], w * msg);
}

// ---------------------------------------------------------------------------
// Cross-type mean + ELU:  G = elu( agg / max(flag1+flag2, 1) )
// flag_i derived from den_i[node*8] > 0 (exp terms are strictly positive).
// ---------------------------------------------------------------------------
__global__ __launch_bounds__(256)
void prep_update_kernel(const float* __restrict__ agg, const float* __restrict__ den1,
                        const float* __restrict__ den2, float* __restrict__ G, int n) {
    long long i = (long long)blockIdx.x * blockDim.x + threadIdx.x;
    if (i >= (long long)n * DIM) return;
    int node = (int)(i >> 8);
    float f1 = (den1[(size_t)node * NHEAD] > 0.0f) ? 1.0f : 0.0f;
    float f2 = den2 ? ((den2[(size_t)node * NHEAD] > 0.0f) ? 1.0f : 0.0f) : 0.0f;
    float x = agg[i] / fmaxf(f1 + f2, 1.0f);
    G[i] = (x > 0.0f) ? x : (__expf(x) - 1.0f);   // ELU(alpha=1)
}

// ---------------------------------------------------------------------------
// Gated residual + LayerNorm.  One 256-thread block per node row.
// ---------------------------------------------------------------------------
__global__ __launch_bounds__(256)
void finalize_kernel(const float* __restrict__ X, const float* __restrict__ H,
                     const float* __restrict__ g, const float* __restrict__ b,
                     const float* __restrict__ res_alpha, int aidx,
                     float* __restrict__ out, int n) {
    __shared__ float red[8];
    __shared__ float stat[2];
    const int node = blockIdx.x;
    const int j = threadIdx.x;
    const float alpha = 1.0f / (1.0f + __expf(-res_alpha[aidx]));
    const float u = alpha * X[(size_t)node * DIM + j] +
                    (1.0f - alpha) * H[(size_t)node * DIM + j];
    // mean
    float s = u;
    for (int off = 16; off > 0; off >>= 1) s += __shfl_xor(s, off, 32);
    if ((j & 31) == 0) red[j >> 5] = s;
    __syncthreads();
    if (j == 0) {
        float t = 0.0f;
        for (int w = 0; w < 8; ++w) t += red[w];
        stat[0] = t * (1.0f / DIM);
    }
    __syncthreads();
    const float mean = stat[0];
    const float dv = u - mean;
    // variance (ddof=0)
    float s2 = dv * dv;
    for (int off = 16; off > 0; off >>= 1) s2 += __shfl_xor(s2, off, 32);
    __syncthreads();
    if ((j & 31) == 0) red[j >> 5] = s2;
    __syncthreads();
    if (j == 0) {
        float t = 0.0f;
        for (int w = 0; w < 8; ++w) t += red[w];
        stat[1] = t * (1.0f / DIM);
    }
    __syncthreads();
    out[(size_t)node * DIM + j] = dv * rsqrtf(stat[1] + 1e-5f) * g[j] + b[j];
}

// ---------------------------------------------------------------------------
// Host launch
// ---------------------------------------------------------------------------
extern "C" void kernel_launch(void* const* d_in, const int* in_sizes, int n_in,
                              void* d_out, int out_size, void* d_ws, size_t ws_size,
                              hipStream_t stream) {
    const float* Ha   = (const float*)d_in[0];
    const float* Hb   = (const float*)d_in[1];
    const float* emb  = (const float*)d_in[2];
    const float* Wq   = (const float*)d_in[3];
    const float* Wk   = (const float*)d_in[4];
    const float* Wm   = (const float*)d_in[5];
    const float* Wa   = (const float*)d_in[6];
    const float* rteW = (const float*)d_in[7];
    const float* rteb = (const float*)d_in[8];
    const float* ln_g = (const float*)d_in[9];
    const float* ln_b = (const float*)d_in[10];
    const float* ralp = (const float*)d_in[11];
    const float* Watt = (const float*)d_in[12];
    const float* Wmsg = (const float*)d_in[13];
    const float* prior= (const float*)d_in[14];
    const int* srcp[3] = {(const int*)d_in[15], (const int*)d_in[18], (const int*)d_in[21]};
    const int* dstp[3] = {(const int*)d_in[16], (const int*)d_in[19], (const int*)d_in[22]};
    const int* ttp[3]  = {(const int*)d_in[17], (const int*)d_in[20], (const int*)d_in[23]};

    const int NA = in_sizes[0] / DIM;
    const int NB = in_sizes[1] / DIM;
    const int ML = in_sizes[2] / DIM;         // 27759
    const int Ecnt[3] = {in_sizes[15], in_sizes[18], in_sizes[21]};
    int Emax = Ecnt[0];
    if (Ecnt[1] > Emax) Emax = Ecnt[1];
    if (Ecnt[2] > Emax) Emax = Ecnt[2];
    const int WW = DIM * DIM;                 // 65536

    // ---- workspace layout (floats) ----
    float* ws = (float*)d_ws;
    size_t off = 0;
    auto alloc = [&](size_t nf) { float* p = ws + off; off += nf; return p; };
    float* WkF  = alloc((size_t)3 * WW);
    float* WmF  = alloc((size_t)3 * WW);
    float* RkF  = alloc((size_t)3 * WW);
    float* RmF  = alloc((size_t)3 * WW);
    float* bkF  = alloc(3 * DIM);
    float* bmF  = alloc(3 * DIM);
    float* Qa   = alloc((size_t)NA * DIM);    // later reused as Xa
    float* Qb   = alloc((size_t)NB * DIM);    // later reused as Xb
    float* KW   = alloc((size_t)NA * DIM);    // later reused as Ga
    float* MW   = alloc((size_t)NA * DIM);    // later reused as Gb
    float* KR   = alloc((size_t)ML * DIM);
    float* MR   = alloc((size_t)ML * DIM);
    float* att  = alloc((size_t)Emax * NHEAD);
    float* amaxf= alloc((size_t)NA * NHEAD);  // ordered-uint encoded; 0 == "-inf"
    float* den1 = alloc((size_t)NA * NHEAD);
    float* den2 = alloc((size_t)NA * NHEAD);
    float* den3 = alloc((size_t)NB * NHEAD);
    float* agg_a= alloc((size_t)NA * DIM);
    float* agg_b= alloc((size_t)NB * DIM);
    unsigned* amax = (unsigned*)amaxf;
    (void)ws_size; (void)n_in; (void)out_size;

    auto zero = [&](float* p, long long n) {
        zero_kernel<<<dim3((unsigned)((n + 255) / 256)), dim3(256), 0, stream>>>(p, n);
    };
    auto gemm = [&](const float* IN, const float* W, const float* bias, float* OUT, int rows) {
        gemm256_wmma_kernel<<<dim3((unsigned)((rows + 31) / 32)), dim3(512), 0, stream>>>(
            IN, W, bias, OUT, rows);
    };

    // ---- stage A: fold Watt/Wmsg + RTE linear into projection weights ----
    for (int e = 0; e < 3; ++e) {
        const int s = (e == 1) ? 1 : 0;       // source node type per edge type
        fold_head_kernel<<<dim3(256), dim3(256), 0, stream>>>(
            Wk + (size_t)s * WW, Watt + (size_t)e * DHEAD * DHEAD, WkF + (size_t)e * WW);
        fold_head_kernel<<<dim3(256), dim3(256), 0, stream>>>(
            Wm + (size_t)s * WW, Wmsg + (size_t)e * DHEAD * DHEAD, WmF + (size_t)e * WW);
        fold_rte_kernel<<<dim3(256), dim3(256), 0, stream>>>(
            WkF + (size_t)e * WW, rteW + (size_t)s * WW, rteb + (size_t)s * DIM,
            RkF + (size_t)e * WW, bkF + (size_t)e * DIM);
        fold_rte_kernel<<<dim3(256), dim3(256), 0, stream>>>(
            WmF + (size_t)e * WW, rteW + (size_t)s * WW, rteb + (size_t)s * DIM,
            RmF + (size_t)e * WW, bmF + (size_t)e * DIM);
    }

    // ---- stage B: Q projections (WMMA) ----
    gemm(Ha, Wq, nullptr, Qa, NA);
    gemm(Hb, Wq + WW, nullptr, Qb, NB);

    zero(agg_a, (long long)NA * DIM);
    zero(agg_b, (long long)NB * DIM);
    zero(den1, (long long)NA * NHEAD);
    zero(den2, (long long)NA * NHEAD);
    zero(den3, (long long)NB * NHEAD);

    // ---- stage C: per edge type, tables (WMMA) + three scatter passes ----
    for (int e = 0; e < 3; ++e) {
        const float* Hsrc = (e == 1) ? Hb : Ha;
        const int    nsrc = (e == 1) ? NB : NA;
        const float* Qd   = (e == 2) ? Qb : Qa;
        const int    ndst = (e == 2) ? NB : NA;
        float* den_e = (e == 0) ? den1 : (e == 1) ? den2 : den3;
        float* agg_e = (e == 2) ? agg_b : agg_a;
        const int ne = Ecnt[e];

        gemm(Hsrc, WkF + (size_t)e * WW, nullptr, KW, nsrc);
        gemm(Hsrc, WmF + (size_t)e * WW, nullptr, MW, nsrc);
        gemm(emb, RkF + (size_t)e * WW, bkF + (size_t)e * DIM, KR, ML);
        gemm(emb, RmF + (size_t)e * WW, bmF + (size_t)e * DIM, MR, ML);

        zero(amaxf, (long long)ndst * NHEAD);   // encoded 0 < any finite value

        edge_att_kernel<<<dim3((unsigned)((ne + 7) / 8)), dim3(256), 0, stream>>>(
            KW, KR, Qd, srcp[e], dstp[e], ttp[e], att, amax, prior, e, ML, ne);
        edge_softmax_kernel<<<dim3((unsigned)(((long long)ne * NHEAD + 255) / 256)),
                              dim3(256), 0, stream>>>(dstp[e], amax, att, den_e, ne);
        edge_scatter_kernel<<<dim3((unsigned)ne), dim3(256), 0, stream>>>(
            MW, MR, srcp[e], dstp[e], ttp[e], att, den_e, agg_e, ML, ne);
    }

    // ---- stage D: cross-type mean + ELU, output projection (WMMA), LN ----
    float* Ga = KW;   // KW/MW dead after edge passes
    float* Gb = MW;
    float* Xa = Qa;   // Qa/Qb dead after pass 1
    float* Xb = Qb;
    prep_update_kernel<<<dim3((unsigned)(((long long)NA * DIM + 255) / 256)),
                         dim3(256), 0, stream>>>(agg_a, den1, den2, Ga, NA);
    prep_update_kernel<<<dim3((unsigned)(((long long)NB * DIM + 255) / 256)),
                         dim3(256), 0, stream>>>(agg_b, den3, nullptr, Gb, NB);
    gemm(Ga, Wa, nullptr, Xa, NA);
    gemm(Gb, Wa + WW, nullptr, Xb, NB);

    float* out = (float*)d_out;
    finalize_kernel<<<dim3((unsigned)NA), dim3(256), 0, stream>>>(
        Xa, Ha, ln_g, ln_b, ralp, 0, out, NA);
    finalize_kernel<<<dim3((unsigned)NB), dim3(256), 0, stream>>>(
        Xb, Hb, ln_g + DIM, ln_b + DIM, ralp, 1, out + (size_t)NA * DIM, NB);
}